// EulerCTRNN_8169027797237
// MI455X (gfx1250) — compile-verified
//
#include <hip/hip_runtime.h>
#include <hip/hip_bf16.h>

// EulerCTRNN fused scan for gfx1250 (MI455X).
// 8 persistent workgroups (one 16-batch-row slice each), 512 threads = 16 waves.
// Per step: r = tanh(h) lives in LDS; each wave computes two 16x16 output tiles
// of drive = r@Wh + x_t@Wi via V_WMMA_F32_16X16X4_F32 (fp32 matrix pipe, exact).
// Weights are pre-packed into a WMMA-fragment-interleaved layout in d_ws so the
// B operand is one coalesced global_load_b128 per 2 K-blocks per tile. The x_t
// tile is staged with GLOBAL_LOAD_ASYNC_TO_LDS_B128 (ASYNCcnt-tracked).

#define ALPHA_C  0.1f
#define NOISE_C  0.1f

constexpr int Bc = 128;
constexpr int Tc = 512;
constexpr int Ic = 128;
constexpr int Hc = 512;

// padded LDS row strides: 516 % 64 == 4, 132 % 64 == 4 -> conflict-free
// column reads when 16 lanes each read a different row at the same column.
constexpr int RS = 516;
constexpr int XS = 132;

// packed-weight sizes (floats): layout [K/8][H][2 halves][4]
constexpr size_t WHP_ELEMS = (size_t)(Hc / 8) * Hc * 8;   // 262144
constexpr size_t WIP_ELEMS = (size_t)(Ic / 8) * Hc * 8;   //  65536
constexpr size_t WS_NEEDED = (WHP_ELEMS + WIP_ELEMS) * sizeof(float);

typedef float v2f __attribute__((ext_vector_type(2)));
typedef float v8f __attribute__((ext_vector_type(8)));
typedef int   vi4 __attribute__((vector_size(4 * sizeof(int))));

#if defined(__has_builtin)
#if __has_builtin(__builtin_amdgcn_global_load_async_to_lds_b128)
#define HAVE_ASYNC_LDS 1
#endif
#if __has_builtin(__builtin_amdgcn_s_wait_asynccnt)
#define HAVE_WAIT_ASYNC 1
#endif
#endif

__device__ __forceinline__ void wait_asynccnt0() {
#if defined(HAVE_WAIT_ASYNC)
    __builtin_amdgcn_s_wait_asynccnt(0);
#elif defined(HAVE_ASYNC_LDS)
    asm volatile("s_wait_asynccnt 0x0" ::: "memory");
#endif
}

// Re-pack a [K,H] row-major weight matrix into fragment-interleaved form:
// dst[(((kb2*H + n)*2 + half)*4) + j] = src[(8*kb2 + 2*half + (j&1) + ((j>>1)*4)) * H + n]
// so lane l (half = l>>4, n = tile_n0 + (l&15)) loads one float4 that supplies
// the B fragments of K-blocks 2*kb2 and 2*kb2+1.
__global__ void pack_weights_kernel(const float* __restrict__ src,
                                    float* __restrict__ dst,
                                    int K, int Hdim)
{
    const int total = (K / 8) * Hdim * 8;
    for (int i = blockIdx.x * blockDim.x + threadIdx.x; i < total;
         i += gridDim.x * blockDim.x) {
        const int j    = i & 3;
        const int half = (i >> 2) & 1;
        const int n    = (i >> 3) % Hdim;
        const int kb2  = (i >> 3) / Hdim;
        const int row  = 8 * kb2 + 2 * half + (j & 1) + ((j >> 1) << 2);
        dst[i] = src[(size_t)row * Hdim + n];
    }
}

template <bool PACKED>
__launch_bounds__(512, 1)
__global__ void ctrnn_scan_kernel(const float* __restrict__ x,
                                  const float* __restrict__ noise,
                                  const float* __restrict__ Wh,   // raw [H,H]
                                  const float* __restrict__ bh,
                                  const float* __restrict__ Wi,   // raw [I,H]
                                  const float* __restrict__ Wo,
                                  const float* __restrict__ bo,
                                  const float* __restrict__ WhP,  // packed (PACKED only)
                                  const float* __restrict__ WiP,  // packed (PACKED only)
                                  float* __restrict__ z_out,
                                  float* __restrict__ rates_out)
{
    __shared__ float h_lds[16 * RS];    // hidden state, 16 rows x 512 (padded)
    __shared__ float r_lds[16 * RS];    // tanh(h), A-operand of recurrent GEMM
    __shared__ float xt_lds[16 * XS];   // x_t tile, A-operand of input GEMM

    const int tid  = threadIdx.x;
    const int w    = tid >> 5;         // wave id, 0..15
    const int l    = tid & 31;         // lane id
    const int m    = l & 15;           // A-fragment row (M)
    const int half = l >> 4;           // fragment K-half select
    const int khi  = half << 1;        // K offset within a K-block: 0 or 2
    const int nl   = l & 15;           // B/C/D fragment column (N)
    const int b0   = blockIdx.x * 16;  // first batch row of this slice

    // two N-tiles per wave: columns [32w, 32w+16) and [32w+16, 32w+32)
    const int ca = (2 * w) * 16 + nl;
    const int cb = (2 * w + 1) * 16 + nl;

    // h0 = 0, r0 = tanh(0) = 0
    for (int i = tid; i < 16 * RS; i += 512) { h_lds[i] = 0.0f; r_lds[i] = 0.0f; }
    __syncthreads();

    for (int t = 0; t < Tc; ++t) {
        // ---- stage x_t tile (16 x 128) into LDS ----
#if defined(HAVE_ASYNC_LDS)
        {
            // one 16B chunk per thread: 16 rows x 32 chunks
            const int row = tid >> 5;
            const int cc4 = (tid & 31) << 2;
            const float* gp = &x[((size_t)(b0 + row) * Tc + t) * Ic + cc4];
            float*       lp = &xt_lds[row * XS + cc4];
            __builtin_amdgcn_global_load_async_to_lds_b128(
                (__attribute__((address_space(1))) vi4*)(uintptr_t)(const void*)gp,
                (__attribute__((address_space(3))) vi4*)lp,
                0, 0);
        }
#else
        #pragma unroll
        for (int i = 0; i < 4; ++i) {
            int idx = tid + 512 * i;
            int row = idx >> 7, cc = idx & 127;
            xt_lds[row * XS + cc] = x[((size_t)(b0 + row) * Tc + t) * Ic + cc];
        }
#endif
        if (t + 1 < Tc) {
            int row = tid >> 7, cc = tid & 127;
            __builtin_prefetch(&x[((size_t)(b0 + row) * Tc + (t + 1)) * Ic + cc], 0, 0);
        }
        wait_asynccnt0();
        __syncthreads();

        // ---- drive = tanh(h) @ Wh  +  x_t @ Wi  (two 16x16 tiles per wave) ----
        v8f acc0 = {};
        v8f acc1 = {};

        if (PACKED) {
            // recurrent part: K = 512, two K-blocks (K=8) per packed b128 load
            for (int kb2 = 0; kb2 < Hc / 8; ++kb2) {
                const int k0 = kb2 * 8;
                v2f a0, a1;
                a0.x = r_lds[m * RS + k0 + khi];
                a0.y = r_lds[m * RS + k0 + khi + 1];
                a1.x = r_lds[m * RS + k0 + 4 + khi];
                a1.y = r_lds[m * RS + k0 + 4 + khi + 1];
                const float4 pA = *reinterpret_cast<const float4*>(
                    WhP + ((((size_t)kb2 * Hc + ca) << 1) + half) * 4);
                const float4 pB = *reinterpret_cast<const float4*>(
                    WhP + ((((size_t)kb2 * Hc + cb) << 1) + half) * 4);
                v2f bA0; bA0.x = pA.x; bA0.y = pA.y;
                v2f bA1; bA1.x = pA.z; bA1.y = pA.w;
                v2f bB0; bB0.x = pB.x; bB0.y = pB.y;
                v2f bB1; bB1.x = pB.z; bB1.y = pB.w;
                acc0 = __builtin_amdgcn_wmma_f32_16x16x4_f32(false, a0, false, bA0,
                                                             (short)0, acc0, false, false);
                acc1 = __builtin_amdgcn_wmma_f32_16x16x4_f32(false, a0, false, bB0,
                                                             (short)0, acc1, false, false);
                acc0 = __builtin_amdgcn_wmma_f32_16x16x4_f32(false, a1, false, bA1,
                                                             (short)0, acc0, false, false);
                acc1 = __builtin_amdgcn_wmma_f32_16x16x4_f32(false, a1, false, bB1,
                                                             (short)0, acc1, false, false);
            }
            // input part: K = 128
            for (int kb2 = 0; kb2 < Ic / 8; ++kb2) {
                const int k0 = kb2 * 8;
                v2f a0, a1;
                a0.x = xt_lds[m * XS + k0 + khi];
                a0.y = xt_lds[m * XS + k0 + khi + 1];
                a1.x = xt_lds[m * XS + k0 + 4 + khi];
                a1.y = xt_lds[m * XS + k0 + 4 + khi + 1];
                const float4 pA = *reinterpret_cast<const float4*>(
                    WiP + ((((size_t)kb2 * Hc + ca) << 1) + half) * 4);
                const float4 pB = *reinterpret_cast<const float4*>(
                    WiP + ((((size_t)kb2 * Hc + cb) << 1) + half) * 4);
                v2f bA0; bA0.x = pA.x; bA0.y = pA.y;
                v2f bA1; bA1.x = pA.z; bA1.y = pA.w;
                v2f bB0; bB0.x = pB.x; bB0.y = pB.y;
                v2f bB1; bB1.x = pB.z; bB1.y = pB.w;
                acc0 = __builtin_amdgcn_wmma_f32_16x16x4_f32(false, a0, false, bA0,
                                                             (short)0, acc0, false, false);
                acc1 = __builtin_amdgcn_wmma_f32_16x16x4_f32(false, a0, false, bB0,
                                                             (short)0, acc1, false, false);
                acc0 = __builtin_amdgcn_wmma_f32_16x16x4_f32(false, a1, false, bA1,
                                                             (short)0, acc0, false, false);
                acc1 = __builtin_amdgcn_wmma_f32_16x16x4_f32(false, a1, false, bB1,
                                                             (short)0, acc1, false, false);
            }
        } else {
            // fallback: direct (unpacked) weight loads
            for (int kb = 0; kb < Hc / 4; ++kb) {
                const int k0 = kb * 4;
                v2f a;
                a.x = r_lds[m * RS + k0 + khi];
                a.y = r_lds[m * RS + k0 + khi + 1];
                v2f bA, bB;
                bA.x = Wh[(size_t)(k0 + khi)     * Hc + ca];
                bA.y = Wh[(size_t)(k0 + khi + 1) * Hc + ca];
                bB.x = Wh[(size_t)(k0 + khi)     * Hc + cb];
                bB.y = Wh[(size_t)(k0 + khi + 1) * Hc + cb];
                acc0 = __builtin_amdgcn_wmma_f32_16x16x4_f32(false, a, false, bA,
                                                             (short)0, acc0, false, false);
                acc1 = __builtin_amdgcn_wmma_f32_16x16x4_f32(false, a, false, bB,
                                                             (short)0, acc1, false, false);
            }
            for (int kb = 0; kb < Ic / 4; ++kb) {
                const int k0 = kb * 4;
                v2f a;
                a.x = xt_lds[m * XS + k0 + khi];
                a.y = xt_lds[m * XS + k0 + khi + 1];
                v2f bA, bB;
                bA.x = Wi[(size_t)(k0 + khi)     * Hc + ca];
                bA.y = Wi[(size_t)(k0 + khi + 1) * Hc + ca];
                bB.x = Wi[(size_t)(k0 + khi)     * Hc + cb];
                bB.y = Wi[(size_t)(k0 + khi + 1) * Hc + cb];
                acc0 = __builtin_amdgcn_wmma_f32_16x16x4_f32(false, a, false, bA,
                                                             (short)0, acc0, false, false);
                acc1 = __builtin_amdgcn_wmma_f32_16x16x4_f32(false, a, false, bB,
                                                             (short)0, acc1, false, false);
            }
        }
        __syncthreads();   // all reads of r_lds for step t are done

        // ---- Euler update + rates store (C/D layout: M = e + 8*(l>>4), N = l&15) ----
        #pragma unroll
        for (int e = 0; e < 8; ++e) {
            const int Me = e + (half << 3);
            const int gr = b0 + Me;
            const size_t base = ((size_t)gr * Tc + t) * Hc;
            {
                float drive = acc0[e] + bh[ca] + NOISE_C * noise[base + ca];
                float hold  = h_lds[Me * RS + ca];
                float hnew  = (1.0f - ALPHA_C) * hold + ALPHA_C * drive;
                h_lds[Me * RS + ca] = hnew;
                float rt = tanhf(hnew);
                r_lds[Me * RS + ca] = rt;
                rates_out[base + ca] = rt;
            }
            {
                float drive = acc1[e] + bh[cb] + NOISE_C * noise[base + cb];
                float hold  = h_lds[Me * RS + cb];
                float hnew  = (1.0f - ALPHA_C) * hold + ALPHA_C * drive;
                h_lds[Me * RS + cb] = hnew;
                float rt = tanhf(hnew);
                r_lds[Me * RS + cb] = rt;
                rates_out[base + cb] = rt;
            }
        }
        __syncthreads();   // r_lds now holds rates(t)

        // ---- readout z[b,t] = rates . Wo + bo : wave w reduces batch row w ----
        float s = 0.0f;
        #pragma unroll
        for (int i = 0; i < Hc / 32; ++i) {
            const int col = l + 32 * i;
            s += r_lds[w * RS + col] * Wo[col];
        }
        #pragma unroll
        for (int off = 16; off > 0; off >>= 1) s += __shfl_xor(s, off, 32);
        if (l == 0) z_out[(size_t)(b0 + w) * Tc + t] = s + bo[0];
        __syncthreads();   // protect r_lds reads from next step's writes
    }
}

extern "C" void kernel_launch(void* const* d_in, const int* in_sizes, int n_in,
                              void* d_out, int out_size, void* d_ws, size_t ws_size,
                              hipStream_t stream) {
    const float* x     = (const float*)d_in[0];   // [B,T,I]
    const float* noise = (const float*)d_in[1];   // [B,T,H]
    const float* Wh    = (const float*)d_in[2];   // [H,H]
    const float* bh    = (const float*)d_in[3];   // [H]
    const float* Wi    = (const float*)d_in[4];   // [I,H]
    const float* Wo    = (const float*)d_in[5];   // [H,1]
    const float* bo    = (const float*)d_in[6];   // [1]

    float* z_out     = (float*)d_out;                     // [B,T] flat
    float* rates_out = (float*)d_out + (size_t)Bc * Tc;   // [B,T,H]

    dim3 grid(Bc / 16);   // 8 persistent workgroups
    dim3 block(512);      // 16 waves (wave32)

    if (ws_size >= WS_NEEDED) {
        float* WhP = (float*)d_ws;
        float* WiP = (float*)d_ws + WHP_ELEMS;
        pack_weights_kernel<<<512, 256, 0, stream>>>(Wh, WhP, Hc, Hc);
        pack_weights_kernel<<<128, 256, 0, stream>>>(Wi, WiP, Ic, Hc);
        ctrnn_scan_kernel<true><<<grid, block, 0, stream>>>(
            x, noise, Wh, bh, Wi, Wo, bo, WhP, WiP, z_out, rates_out);
    } else {
        ctrnn_scan_kernel<false><<<grid, block, 0, stream>>>(
            x, noise, Wh, bh, Wi, Wo, bo, nullptr, nullptr, z_out, rates_out);
    }
}